// PaiConvDG_10934986736326
// MI455X (gfx1250) — compile-verified
//
#include <hip/hip_runtime.h>
#include <hip/hip_bf16.h>

// ---- problem constants (from reference) ----
#define BQ     8
#define NPTS   4096
#define NP     (BQ * NPTS)   // 32768 points
#define FEATS  128
#define KN     20            // spiral neighbors
#define KSN    20            // permatrix output cols
#define KPAD   32            // padded column count (2 x 16-wide WMMA tiles)
#define OUTC   64
#define WPAD   (FEATS + 4)   // padded row stride for W in LDS
#define SPAD   (FEATS + 4)   // padded column stride for Sflat in LDS
#define PSTR   24            // row stride of transposed permatrix in LDS
#define PPB    8             // points per workgroup

typedef __attribute__((ext_vector_type(2))) float v2f_t;
typedef __attribute__((ext_vector_type(8))) float v8f_t;
typedef int v4i_gcc __attribute__((vector_size(16)));   // matches builtin param type

// ---- async global->LDS copy (CDNA5 ASYNCcnt path), with safe fallback ----
#if defined(__gfx1250__) && __has_builtin(__builtin_amdgcn_global_load_async_to_lds_b128)
#define ASYNC_LDS 1
#else
#define ASYNC_LDS 0
#endif

__device__ __forceinline__ void copy16_to_lds(const float* __restrict__ gsrc,
                                              float* __restrict__ ldst) {
#if ASYNC_LDS
    __builtin_amdgcn_global_load_async_to_lds_b128(
        (__attribute__((address_space(1))) v4i_gcc*)gsrc,
        (__attribute__((address_space(3))) v4i_gcc*)ldst,
        0, 0);
#else
    *(float4*)ldst = *(const float4*)gsrc;
#endif
}

__device__ __forceinline__ void async_lds_wait() {
#if ASYNC_LDS
#if __has_builtin(__builtin_amdgcn_s_wait_asynccnt)
    __builtin_amdgcn_s_wait_asynccnt(0);
#else
    asm volatile("s_wait_asynccnt 0" ::: "memory");
#endif
#endif
}

// ------------------------------------------------------------------
// Phase 0: feature [B][FEATS][N] -> flatT [B*N][FEATS] (tiled transpose)
// ------------------------------------------------------------------
__global__ void transpose_feature(const float* __restrict__ feature,
                                  float* __restrict__ flatT) {
    __shared__ float tile[32][33];
    const int b  = blockIdx.z;
    const int c0 = blockIdx.y * 32;
    const int n0 = blockIdx.x * 32;
    const int tx = threadIdx.x;   // 0..31
    const int ty = threadIdx.y;   // 0..7
#pragma unroll
    for (int i = 0; i < 32; i += 8) {
        const int c = c0 + ty + i;
        tile[ty + i][tx] = feature[((size_t)b * FEATS + c) * NPTS + (n0 + tx)];
    }
    __syncthreads();
#pragma unroll
    for (int i = 0; i < 32; i += 8) {
        const int n = n0 + ty + i;
        flatT[((size_t)b * NPTS + n) * FEATS + (c0 + tx)] = tile[tx][ty + i];
    }
}

// ------------------------------------------------------------------
// Phase 1: fused  S = G^T * P  ->  Y = W * X  -> max+bias
//   one workgroup (256 thr = 8 wave32) handles PPB points; W stays in LDS.
// ------------------------------------------------------------------
__global__ void __launch_bounds__(256)
paiconv_fused(const float* __restrict__ flatT,
              const float* __restrict__ permatrix,
              const float* __restrict__ conv_w,
              const float* __restrict__ conv_b,
              const int*   __restrict__ spirals_index,
              float*       __restrict__ yws) {
    __shared__ float Gs[KN][FEATS];        // gathered neighbor rows (k-major)
    __shared__ float Pt[KPAD][PSTR];       // permatrix TRANSPOSED: Pt[s][k], rows>=20 zero
    __shared__ float Sp[KPAD * SPAD];      // Sflat as padded 128-columns (cols 20..31 zero)
    __shared__ float Ws[OUTC][WPAD];       // conv weight, padded rows
    __shared__ float Ys[OUTC][KPAD];       // conv output tile (cols 20..31 junk)

    const int tid  = threadIdx.x;
    const int wave = tid >> 5;
    const int lane = tid & 31;
    const int g    = lane >> 4;            // lane group (0: lanes 0-15, 1: 16-31)
    const int li   = lane & 15;

    // ---- one-time per block: load W, zero the pad regions ----
    const float4* w4 = (const float4*)conv_w;
    for (int i = tid; i < (OUTC * FEATS) / 4; i += 256) {
        const int o  = i >> 5;             // 32 float4 per 128-float row
        const int d4 = i & 31;
        copy16_to_lds((const float*)&w4[i], &Ws[o][d4 * 4]);
    }
    for (int i = tid; i < (KPAD - KSN) * KN; i += 256)       // Pt pad rows (s>=20)
        Pt[KSN + i / KN][i % KN] = 0.0f;
    for (int i = tid; i < (KPAD - KSN) * FEATS; i += 256)    // Sp pad cols
        Sp[(KSN + (i >> 7)) * SPAD + (i & 127)] = 0.0f;

    for (int pi = 0; pi < PPB; ++pi) {
        const int p = blockIdx.x * PPB + pi;

        // ---- per-point loads ----
        // permatrix row-major [k][s] -> LDS transposed Pt[s][k] (k-contiguous)
        const float* Pg = permatrix + (size_t)p * (KN * KSN);
        for (int i = tid; i < KN * KSN; i += 256) {
            const int k = i / KSN;
            const int s = i % KSN;
            Pt[s][k] = Pg[i];              // coalesced global read
        }

        const int* idxp = spirals_index + (size_t)p * KN;
        for (int i = tid; i < KN * (FEATS / 4); i += 256) {
            const int k   = i >> 5;        // FEATS/4 == 32
            const int c4  = i & 31;
            const int row = idxp[k];
            copy16_to_lds(flatT + (size_t)row * FEATS + c4 * 4, &Gs[k][c4 * 4]);
        }
        if (pi + 1 < PPB) {                // warm GL2 for next point
            __builtin_prefetch(permatrix + (size_t)(p + 1) * (KN * KSN), 0, 0);
            __builtin_prefetch(spirals_index + (size_t)(p + 1) * KN, 0, 0);
        }
        async_lds_wait();
        __syncthreads();

        // ---- GEMM 1: S(128x20) = G^T(128 x K20) * P(20x20), zero-padded N ----
        {
            const int cBase = wave * 16;   // wave -> c-tile
            v8f_t accS0 = {};
            v8f_t accS1 = {};
#pragma unroll
            for (int ks = 0; ks < 5; ++ks) {            // K = 20 = 5 * 4
                const int k0 = ks * 4 + 2 * g;          // VGPR v -> K = 4*ks + v + 2*g
                v2f_t a;                                 // A[M=c][K=k] = Gs[k][c]
                a.x = Gs[k0 + 0][cBase + li];
                a.y = Gs[k0 + 1][cBase + li];
                // B[K=k][N=s] = Pt[s][k]; k-contiguous -> single ds_load_b64 each
                v2f_t b0 = *(const v2f_t*)&Pt[li][k0];        // s-tile 0
                v2f_t b1 = *(const v2f_t*)&Pt[16 + li][k0];   // s-tile 1 (rows>=20 zero)
                accS0 = __builtin_amdgcn_wmma_f32_16x16x4_f32(
                    false, a, false, b0, (short)0, accS0, false, false);
                accS1 = __builtin_amdgcn_wmma_f32_16x16x4_f32(
                    false, a, false, b1, (short)0, accS1, false, false);
            }
            // store S row-major-flat rebucketed into padded 128-columns:
            //   e = c*20 + s ; Sp[(e/128)*SPAD + e%128]
#pragma unroll
            for (int r = 0; r < 8; ++r) {
                const int c = cBase + r + 8 * g;
                const int e = c * KSN + li;
                Sp[(e >> 7) * SPAD + (e & 127)] = accS0[r];
            }
            if (li < KSN - 16) {           // s = 16+li valid only for li < 4
#pragma unroll
                for (int r = 0; r < 8; ++r) {
                    const int c = cBase + r + 8 * g;
                    const int e = c * KSN + 16 + li;
                    Sp[(e >> 7) * SPAD + (e & 127)] = accS1[r];
                }
            }
        }
        __syncthreads();

        // ---- GEMM 2: Y(64x20) = W(64x128) * X(128x20) ----
        // X column a == Sflat[a*128 .. +128) == Sp[a*SPAD .. +128)
        {
            const int oBase = (wave & 3) * 16;
            const int acol  = (wave >> 2) * 16 + li;     // 0..31, pad cols read zero
            const float* wrow = &Ws[oBase + li][0];
            const float* xcol = &Sp[acol * SPAD];
            v8f_t accY = {};
#pragma unroll 4
            for (int ks = 0; ks < 32; ++ks) {           // K = 128 = 32 * 4
                const int d0 = ks * 4 + 2 * g;
                v2f_t a = *(const v2f_t*)&wrow[d0];      // ds_load_b64
                v2f_t b = *(const v2f_t*)&xcol[d0];      // ds_load_b64
                accY = __builtin_amdgcn_wmma_f32_16x16x4_f32(
                    false, a, false, b, (short)0, accY, false, false);
            }
#pragma unroll
            for (int r = 0; r < 8; ++r)
                Ys[oBase + r + 8 * g][acol] = accY[r];   // unconditional (padded)
        }
        __syncthreads();

        // ---- max over neighbors + bias, write pre-BN channel-major ----
        if (tid < OUTC) {
            float m = -3.402823466e38f;
#pragma unroll
            for (int a = 0; a < KSN; ++a) m = fmaxf(m, Ys[tid][a]);
            yws[(size_t)tid * NP + p] = m + conv_b[tid];
        }
        __syncthreads();                   // protect Gs/Pt/Sp/Ys for next point
    }
}

// ------------------------------------------------------------------
// Phase 2: per-channel batch statistics (mean, 1/sqrt(var+eps))
// ------------------------------------------------------------------
__global__ void bn_stats(const float* __restrict__ yws,
                         float* __restrict__ stats) {
    const int o = blockIdx.x;
    const float* row = yws + (size_t)o * NP;
    float s = 0.0f, s2 = 0.0f;
    for (int i = threadIdx.x; i < NP; i += 256) {
        const float v = row[i];
        s += v; s2 += v * v;
    }
    __shared__ float sh[256], sh2[256];
    sh[threadIdx.x] = s; sh2[threadIdx.x] = s2;
    __syncthreads();
    for (int off = 128; off > 0; off >>= 1) {
        if (threadIdx.x < off) {
            sh[threadIdx.x]  += sh[threadIdx.x + off];
            sh2[threadIdx.x] += sh2[threadIdx.x + off];
        }
        __syncthreads();
    }
    if (threadIdx.x == 0) {
        const float mean = sh[0] * (1.0f / NP);
        const float var  = sh2[0] * (1.0f / NP) - mean * mean;
        stats[o]        = mean;
        stats[OUTC + o] = rsqrtf(var + 1e-5f);
    }
}

// ------------------------------------------------------------------
// Phase 3: normalize + gamma/beta, layout [B][OUTC][N]
// ------------------------------------------------------------------
__global__ void bn_apply(const float* __restrict__ yws,
                         const float* __restrict__ stats,
                         const float* __restrict__ gamma,
                         const float* __restrict__ beta,
                         float* __restrict__ out) {
    const int idx = blockIdx.x * blockDim.x + threadIdx.x;   // o*NP + p
    if (idx >= OUTC * NP) return;
    const int o  = idx / NP;
    const int pp = idx - o * NP;
    const int b  = pp / NPTS;
    const int n  = pp - b * NPTS;
    const float v = yws[idx];
    const float r = (v - stats[o]) * stats[OUTC + o] * gamma[o] + beta[o];
    out[((size_t)b * OUTC + o) * NPTS + n] = r;
}

// ------------------------------------------------------------------
extern "C" void kernel_launch(void* const* d_in, const int* in_sizes, int n_in,
                              void* d_out, int out_size, void* d_ws, size_t ws_size,
                              hipStream_t stream) {
    const float* feature   = (const float*)d_in[0];  // [8,128,4096]
    const float* permatrix = (const float*)d_in[1];  // [32768,20,20]
    const float* conv_w    = (const float*)d_in[2];  // [64,128]
    const float* conv_b    = (const float*)d_in[3];  // [64]
    const float* bn_gamma  = (const float*)d_in[4];  // [64]
    const float* bn_beta   = (const float*)d_in[5];  // [64]
    const int*   sidx      = (const int*)d_in[6];    // [32768*20]
    float* out = (float*)d_out;                      // [8,64,4096]

    // workspace carve-out (~25.2 MB total)
    float* flatT = (float*)d_ws;                        // 32768*128
    float* yws   = flatT + (size_t)NP * FEATS;          // 64*32768
    float* stats = yws + (size_t)OUTC * NP;             // 128

    transpose_feature<<<dim3(NPTS / 32, FEATS / 32, BQ), dim3(32, 8), 0, stream>>>(
        feature, flatT);
    paiconv_fused<<<NP / PPB, 256, 0, stream>>>(
        flatT, permatrix, conv_w, conv_b, sidx, yws);
    bn_stats<<<OUTC, 256, 0, stream>>>(yws, stats);
    bn_apply<<<(OUTC * NP + 255) / 256, 256, 0, stream>>>(
        yws, stats, bn_gamma, bn_beta, out);
}